// SparseGemmModel_84344567758946
// MI455X (gfx1250) — compile-verified
//
#include <hip/hip_runtime.h>

// CDNA5 / gfx1250: wave32, WMMA (not MFMA).
typedef __attribute__((ext_vector_type(2))) float v2f;
typedef __attribute__((ext_vector_type(8))) float v8f;

// C(32, N) = A(32, K) * W(N, K)^T, all fp32 row-major.
// One wave computes a 32x32 tile of C via V_WMMA_F32_16X16X4_F32.
// Block = 256 threads = 8 waves -> 256 columns of N per block.
//
// Fragment layouts (CDNA5 ISA 7.12.2, 32-bit data):
//  A (16x4, MxK): lane = M (0..15, repeated in lanes 16..31),
//                 VGPR0/1 = K = {0,1} for lanes 0-15, {2,3} for lanes 16-31.
//  B (4x16, KxN): lane = N, same K-split across lane halves / VGPR pair.
//  C/D (16x16):   lane = N, VGPR v -> M = v (lanes 0-15), M = v+8 (lanes 16-31).
//
// Addressing: uniform SGPR base + 32-bit per-lane element offset so the
// backend uses the GVS (saddr+voffset) global-load form instead of 64-bit
// per-lane pointer arithmetic. Weight loads are non-temporal (stream-once,
// 572 MB total > 192 MB L2); activation loads stay temporal (<= 2 MB,
// reused by every N-tile).
__global__ __launch_bounds__(256) void wmma_gemm_xwt(
    const float* __restrict__ A,   // (32, K)
    const float* __restrict__ W,   // (N, K)
    float* __restrict__ C,         // (32, N)
    int K, int N)
{
    const int lane = threadIdx.x & 31;
    const int wave = threadIdx.x >> 5;
    const int r    = lane & 15;          // M or N index within a 16-tile
    const int kh   = (lane >> 4) << 1;   // K half-offset: 0 (lanes 0-15) or 2
    const int n0   = blockIdx.x * 256 + wave * 32;

    // 32-bit element offsets (max 16384*16384 = 2^28, fits signed 32-bit;
    // byte offsets stay below 2^31).
    const int aoff0 = r * K + kh;               // A rows 0..15
    const int aoff1 = (r + 16) * K + kh;        // A rows 16..31
    const int woff0 = (n0 + r) * K + kh;        // W rows n0..n0+15
    const int woff1 = (n0 + 16 + r) * K + kh;   // W rows n0+16..n0+31

    v8f acc00 = {}; v8f acc01 = {}; v8f acc10 = {}; v8f acc11 = {};

    #pragma unroll 8
    for (int k = 0; k < K; k += 4) {
        v2f av0 = *(const v2f*)(A + (aoff0 + k));
        v2f av1 = *(const v2f*)(A + (aoff1 + k));
        v2f bv0 = __builtin_nontemporal_load((const v2f*)(W + (woff0 + k)));
        v2f bv1 = __builtin_nontemporal_load((const v2f*)(W + (woff1 + k)));
        // (neg_a, A, neg_b, B, c_mod, C, reuse_a, reuse_b)
        acc00 = __builtin_amdgcn_wmma_f32_16x16x4_f32(false, av0, false, bv0,
                                                      (short)0, acc00, false, false);
        acc01 = __builtin_amdgcn_wmma_f32_16x16x4_f32(false, av0, false, bv1,
                                                      (short)0, acc01, false, false);
        acc10 = __builtin_amdgcn_wmma_f32_16x16x4_f32(false, av1, false, bv0,
                                                      (short)0, acc10, false, false);
        acc11 = __builtin_amdgcn_wmma_f32_16x16x4_f32(false, av1, false, bv1,
                                                      (short)0, acc11, false, false);
    }

    // Write back: VGPR v holds M = v + 8*(lane>>4); N = n0 + (lane&15) (+16 for hi tile).
    const int mb   = (lane >> 4) * 8;
    const int n_lo = n0 + r;
    const int n_hi = n0 + 16 + r;
    #pragma unroll
    for (int v = 0; v < 8; ++v) {
        C[(mb + v) * N + n_lo]        = acc00[v];
        C[(mb + v) * N + n_hi]        = acc01[v];
        C[(16 + mb + v) * N + n_lo]   = acc10[v];
        C[(16 + mb + v) * N + n_hi]   = acc11[v];
    }
}

extern "C" void kernel_launch(void* const* d_in, const int* in_sizes, int n_in,
                              void* d_out, int out_size, void* d_ws, size_t ws_size,
                              hipStream_t stream) {
    (void)in_sizes; (void)n_in; (void)out_size; (void)ws_size;

    const float* x  = (const float*)d_in[0];  // (32, 256)
    const float* w1 = (const float*)d_in[1];  // (1024, 256)
    const float* w2 = (const float*)d_in[2];  // (4096, 1024)
    const float* w3 = (const float*)d_in[3];  // (16384, 4096)
    const float* w4 = (const float*)d_in[4];  // (4096, 16384)
    const float* w5 = (const float*)d_in[5];  // (1024, 4096)
    const float* w6 = (const float*)d_in[6];  // (256, 1024)

    // Ping-pong activation buffers in workspace.
    // act0 max: 32*16384 fp32 = 2 MB; act1 max: 32*4096 fp32 = 0.5 MB.
    float* act0 = (float*)d_ws;
    float* act1 = (float*)((char*)d_ws + (size_t)32 * 16384 * sizeof(float));

    dim3 blk(256);
    wmma_gemm_xwt<<<dim3(1024  / 256), blk, 0, stream>>>(x,    w1, act0, 256,   1024);
    wmma_gemm_xwt<<<dim3(4096  / 256), blk, 0, stream>>>(act0, w2, act1, 1024,  4096);
    wmma_gemm_xwt<<<dim3(16384 / 256), blk, 0, stream>>>(act1, w3, act0, 4096,  16384);
    wmma_gemm_xwt<<<dim3(4096  / 256), blk, 0, stream>>>(act0, w4, act1, 16384, 4096);
    wmma_gemm_xwt<<<dim3(1024  / 256), blk, 0, stream>>>(act1, w5, act0, 4096,  1024);
    wmma_gemm_xwt<<<dim3(256   / 256), blk, 0, stream>>>(act0, w6, (float*)d_out, 1024, 256);
}